// LATTE_61168924230218
// MI455X (gfx1250) — compile-verified
//
#include <hip/hip_runtime.h>
#include <hip/hip_bf16.h>
#include <math.h>

// Problem constants (from reference)
#define N_NODES 50000
#define DIM     128
#define NH      4
#define CH      32
#define NE      800000

typedef float v2f __attribute__((ext_vector_type(2)));
typedef float v8f __attribute__((ext_vector_type(8)));

// ---------------------------------------------------------------------------
// CDNA5 async global->LDS copy (tracked by ASYNCcnt). Per ISA 15.18.3:
//   global_load_async_to_lds_b128 vdst, vaddr, off
// vdst = VGPR holding LDS byte address (low 32 bits of generic LDS pointer),
// vaddr = VGPR pair holding 64-bit global address.
// ---------------------------------------------------------------------------
__device__ __forceinline__ void async_copy_b128(const float* gsrc, float* ldst) {
    unsigned lds_off = (unsigned)(size_t)ldst;             // LDS aperture: addr[31:0]
    unsigned long long ga = (unsigned long long)(size_t)gsrc;
    asm volatile("global_load_async_to_lds_b128 %0, %1, off"
                 :: "v"(lds_off), "v"(ga) : "memory");
}
#define WAIT_ASYNC_0() asm volatile("s_wait_asynccnt 0x0" ::: "memory")
#define WAIT_ASYNC_2() asm volatile("s_wait_asynccnt 0x2" ::: "memory")

// ---------------------------------------------------------------------------
// float atomic-max via signed/unsigned integer ordering trick
// ---------------------------------------------------------------------------
__device__ __forceinline__ void atomicMaxF(float* addr, float val) {
    if (val >= 0.0f) atomicMax((int*)addr, __float_as_int(val));
    else             atomicMin((unsigned int*)addr, __float_as_uint(val));
}

// ---------------------------------------------------------------------------
// Kernel 1: fused dual GEMM  l = x@Wl + bl ; r = x@Wr + br  (fp32 WMMA)
// block = 256 threads (8 waves). blockIdx.y selects Wl/Wr. Each block caches
// its W (128x128 = 64KB) in LDS via async-to-LDS, then loops over 16-row node
// tiles with a double-buffered async pipeline; each wave owns one 16x16
// output tile (8 waves * 16 = 128 cols).
// WMMA layouts per ISA 7.12.2:
//   A 16x4 f32 : lanes0-15 M=lane, VGPR0=K+0, VGPR1=K+1 ; lanes16-31 K+2/K+3
//   B 4x16 f32 : lanes0-15 N=lane, VGPR0=K+0, VGPR1=K+1 ; lanes16-31 K+2/K+3
//   C 16x16 f32: VGPRv -> (M=v, N=lane) lanes0-15 ; (M=v+8, N=lane-16)
// ---------------------------------------------------------------------------
__global__ __launch_bounds__(256)
void LATTE_gemm(const float* __restrict__ x,
                const float* __restrict__ Wl, const float* __restrict__ bl,
                const float* __restrict__ Wr, const float* __restrict__ br,
                float* __restrict__ lbuf, float* __restrict__ rbuf) {
    extern __shared__ float sm[];
    float* sW  = sm;                     // 128*128 floats (64 KB)
    float* sA0 = sm + DIM * DIM;         // 16*128 floats  (8 KB)
    float* sA1 = sA0 + 16 * DIM;         // 16*128 floats  (8 KB)

    const float* W   = blockIdx.y ? Wr : Wl;
    const float* bs  = blockIdx.y ? br : bl;
    float*       out = blockIdx.y ? rbuf : lbuf;

    const int tid = threadIdx.x;
    const int nTiles = N_NODES / 16;  // 3125 exactly

    // Stage W into LDS asynchronously (16 x b128 per thread).
    for (int i = tid * 4; i < DIM * DIM; i += 256 * 4)
        async_copy_b128(W + i, sW + i);

    // Prologue: async-load first node tile into buffer 0 (2 x b128 per thread).
    int t0 = blockIdx.x;
    for (int i = tid * 4; i < 16 * DIM; i += 256 * 4)
        async_copy_b128(x + (size_t)t0 * 16 * DIM + i, sA0 + i);

    const int lane  = tid & 31;
    const int lmod  = lane & 15;
    const int lhalf = lane >> 4;
    const int col0  = (tid >> 5) * 16;
    const float bv  = bs[col0 + lmod];

    int cur = 0;
    for (int t = t0; t < nTiles; t += gridDim.x) {
        float* sA = cur ? sA1 : sA0;
        float* sN = cur ? sA0 : sA1;

        // Issue async load of the NEXT tile into the other buffer; it overlaps
        // this tile's 32 WMMAs. (Safe: previous iteration's trailing barrier
        // guarantees nobody is still reading sN.)
        const int tn = t + gridDim.x;
        if (tn < nTiles) {
            for (int i = tid * 4; i < 16 * DIM; i += 256 * 4)
                async_copy_b128(x + (size_t)tn * 16 * DIM + i, sN + i);
            WAIT_ASYNC_2();   // async loads retire in order: current tile (+W) done
        } else {
            WAIT_ASYNC_0();
        }
        __syncthreads();      // all waves' transfers for this tile visible

        v8f c0 = {0.f, 0.f, 0.f, 0.f, 0.f, 0.f, 0.f, 0.f};
        v8f c1 = {0.f, 0.f, 0.f, 0.f, 0.f, 0.f, 0.f, 0.f};
#pragma unroll
        for (int k0 = 0; k0 < DIM; k0 += 8) {
            // chain 0
            int ka = k0 + lhalf * 2;
            v2f a0 = *(const v2f*)(sA + lmod * DIM + ka);
            v2f b0;
            b0.x = sW[ka * DIM + col0 + lmod];
            b0.y = sW[(ka + 1) * DIM + col0 + lmod];
            c0 = __builtin_amdgcn_wmma_f32_16x16x4_f32(false, a0, false, b0,
                                                       (short)0, c0, false, false);
            // chain 1 (independent accumulator hides WMMA latency)
            int kb = ka + 4;
            v2f a1 = *(const v2f*)(sA + lmod * DIM + kb);
            v2f b1;
            b1.x = sW[kb * DIM + col0 + lmod];
            b1.y = sW[(kb + 1) * DIM + col0 + lmod];
            c1 = __builtin_amdgcn_wmma_f32_16x16x4_f32(false, a1, false, b1,
                                                       (short)0, c1, false, false);
        }

        const int row0 = t * 16;
        const int mb = lhalf * 8;
#pragma unroll
        for (int v = 0; v < 8; ++v)
            out[(size_t)(row0 + mb + v) * DIM + col0 + lmod] = c0[v] + c1[v] + bv;

        __syncthreads();      // everyone done reading sA before it is re-filled
        cur ^= 1;
    }
}

// ---------------------------------------------------------------------------
// Kernel 2: per-node attention scores + beta softmax. Wave per node.
// lane = channel c; h indexes both head and the per-lane register slot.
// ---------------------------------------------------------------------------
__global__ __launch_bounds__(256)
void LATTE_scores(const float* __restrict__ lbuf, const float* __restrict__ rbuf,
                  const float* __restrict__ attn, const float* __restrict__ Wbeta,
                  const float* __restrict__ bbeta,
                  float* __restrict__ ssrc, float* __restrict__ sdst,
                  float* __restrict__ beta) {
    const int lane = threadIdx.x & 31;
    const int n = blockIdx.x * 8 + (threadIdx.x >> 5);
    if (n >= N_NODES) return;

    float lv[NH], rv[NH], ll[NH], rl[NH];
#pragma unroll
    for (int h = 0; h < NH; ++h) {
        lv[h] = lbuf[(size_t)n * DIM + h * CH + lane];
        rv[h] = rbuf[(size_t)n * DIM + h * CH + lane];
        ll[h] = lv[h] > 0.f ? lv[h] : 0.2f * lv[h];
        rl[h] = rv[h] > 0.f ? rv[h] : 0.2f * rv[h];
    }
#pragma unroll
    for (int m = 0; m < 2; ++m) {
#pragma unroll
        for (int h = 0; h < NH; ++h) {
            // attn shape (2,H,2C): a_i = [:CH], a_j = [CH:]
            float ps = ll[h] * attn[(m * NH + h) * (2 * CH) + CH + lane];
            float pd = rl[h] * attn[(m * NH + h) * (2 * CH) + lane];
#pragma unroll
            for (int off = 16; off > 0; off >>= 1) {
                ps += __shfl_xor(ps, off, 32);
                pd += __shfl_xor(pd, off, 32);
            }
            if (lane == 0) {
                ssrc[((size_t)m * N_NODES + n) * NH + h] = ps;
                sdst[((size_t)m * N_NODES + n) * NH + h] = pd;
            }
        }
    }
    float pb[3];
#pragma unroll
    for (int k = 0; k < 3; ++k) {
        float s = 0.f;
#pragma unroll
        for (int h = 0; h < NH; ++h)
            s += rv[h] * Wbeta[k * DIM + h * CH + lane];
#pragma unroll
        for (int off = 16; off > 0; off >>= 1) s += __shfl_xor(s, off, 32);
        pb[k] = s;
    }
    if (lane == 0) {
        float b0 = pb[0] + bbeta[0], b1 = pb[1] + bbeta[1], b2 = pb[2] + bbeta[2];
        float mx = fmaxf(b0, fmaxf(b1, b2));
        float e0 = expf(b0 - mx), e1 = expf(b1 - mx), e2 = expf(b2 - mx);
        float inv = 1.f / (e0 + e1 + e2);
        beta[(size_t)n * 3 + 0] = e0 * inv;
        beta[(size_t)n * 3 + 1] = e1 * inv;
        beta[(size_t)n * 3 + 2] = e2 * inv;
    }
}

// ---------------------------------------------------------------------------
// Kernel 3: init segment-max to -inf and denom to 0 (deterministic each call)
// ---------------------------------------------------------------------------
__global__ void LATTE_fill(float* __restrict__ m, float* __restrict__ d) {
    int i = blockIdx.x * 256 + threadIdx.x;
    if (i < N_NODES * NH) { m[i] = -INFINITY; d[i] = 0.f; }
}

// ---------------------------------------------------------------------------
// Kernel 4: per-edge segment max (thread per edge, H=4 heads)
// ---------------------------------------------------------------------------
__global__ __launch_bounds__(256)
void LATTE_pass1(const int* __restrict__ ei, const float* __restrict__ ss,
                 const float* __restrict__ sd, const float* __restrict__ sharpen,
                 int rel, float* __restrict__ m) {
    int e = blockIdx.x * 256 + threadIdx.x;
    if (e >= NE) return;
    int s = ei[e], d = ei[NE + e];
    float sh = sharpen[rel];
#pragma unroll
    for (int h = 0; h < NH; ++h) {
        float logit = sh * (sd[d * NH + h] + ss[s * NH + h]);
        atomicMaxF(&m[d * NH + h], logit);
    }
}

// ---------------------------------------------------------------------------
// Kernel 5: per-edge exp-sum (thread per edge)
// ---------------------------------------------------------------------------
__global__ __launch_bounds__(256)
void LATTE_pass2(const int* __restrict__ ei, const float* __restrict__ ss,
                 const float* __restrict__ sd, const float* __restrict__ sharpen,
                 int rel, const float* __restrict__ m, float* __restrict__ denom) {
    int e = blockIdx.x * 256 + threadIdx.x;
    if (e >= NE) return;
    int s = ei[e], d = ei[NE + e];
    float sh = sharpen[rel];
#pragma unroll
    for (int h = 0; h < NH; ++h) {
        float logit = sh * (sd[d * NH + h] + ss[s * NH + h]);
        atomicAdd(&denom[d * NH + h], expf(logit - m[d * NH + h]));
    }
}

// ---------------------------------------------------------------------------
// Kernel 6: message scatter. Wave per edge; lane covers d = 4*lane..4*lane+3
// (head = lane>>3). beta[dst][rel] is folded in here so both relations
// accumulate directly into d_out (saves the emb buffers entirely).
// ---------------------------------------------------------------------------
__global__ __launch_bounds__(256)
void LATTE_pass3(const int* __restrict__ ei, const float* __restrict__ ss,
                 const float* __restrict__ sd, const float* __restrict__ m,
                 const float* __restrict__ denom, const float* __restrict__ lbuf,
                 const float* __restrict__ beta, const float* __restrict__ sharpen,
                 int rel, float* __restrict__ out) {
    const int lane = threadIdx.x & 31;
    const int e = blockIdx.x * 8 + (threadIdx.x >> 5);
    if (e >= NE) return;
    const int s = ei[e], d = ei[NE + e];
    const int h = lane >> 3;
    const float sh = sharpen[rel];
    const float logit = sh * (sd[d * NH + h] + ss[s * NH + h]);
    const float alpha = expf(logit - m[d * NH + h]) / (denom[d * NH + h] + 1e-16f);
    const float w = alpha * beta[(size_t)d * 3 + rel];
    const float4 v = *(const float4*)(lbuf + (size_t)s * DIM + lane * 4);
    float* op = out + (size_t)d * DIM + lane * 4;
    atomicAdd(op + 0, v.x * w);
    atomicAdd(op + 1, v.y * w);
    atomicAdd(op + 2, v.z * w);
    atomicAdd(op + 3, v.w * w);
}

// ---------------------------------------------------------------------------
// Kernel 7: out = relu(out + r * beta[:,2])
// ---------------------------------------------------------------------------
__global__ __launch_bounds__(256)
void LATTE_final(const float* __restrict__ rbuf, const float* __restrict__ beta,
                 float* __restrict__ out) {
    size_t i = (size_t)blockIdx.x * 256 + threadIdx.x;
    if (i >= (size_t)N_NODES * DIM) return;
    size_t n = i >> 7;
    float v = out[i] + rbuf[i] * beta[n * 3 + 2];
    out[i] = v > 0.f ? v : 0.f;
}

// ---------------------------------------------------------------------------
// Host launcher
// Inputs: 0 x, 1 edge_index0, 2 edge_index1, 3 Wl, 4 bl, 5 Wr, 6 br,
//         7 Wbeta, 8 bbeta, 9 attn, 10 sharpen
// Workspace: l, r, ssrc[2], sdst[2], beta, m, denom  (~57 MB)
// ---------------------------------------------------------------------------
extern "C" void kernel_launch(void* const* d_in, const int* in_sizes, int n_in,
                              void* d_out, int out_size, void* d_ws, size_t ws_size,
                              hipStream_t stream) {
    const float* x       = (const float*)d_in[0];
    const int*   ei0     = (const int*)d_in[1];
    const int*   ei1     = (const int*)d_in[2];
    const float* Wl      = (const float*)d_in[3];
    const float* bl      = (const float*)d_in[4];
    const float* Wr      = (const float*)d_in[5];
    const float* br      = (const float*)d_in[6];
    const float* Wbeta   = (const float*)d_in[7];
    const float* bbeta   = (const float*)d_in[8];
    const float* attn    = (const float*)d_in[9];
    const float* sharpen = (const float*)d_in[10];
    float* out = (float*)d_out;

    float* ws   = (float*)d_ws;
    float* lbuf = ws;                                  // N*D
    float* rbuf = lbuf + (size_t)N_NODES * DIM;        // N*D
    float* ssrc = rbuf + (size_t)N_NODES * DIM;        // 2*N*H
    float* sdst = ssrc + (size_t)2 * N_NODES * NH;     // 2*N*H
    float* beta = sdst + (size_t)2 * N_NODES * NH;     // N*3
    float* mbuf = beta + (size_t)N_NODES * 3;          // N*H
    float* dbuf = mbuf + (size_t)N_NODES * NH;         // N*H

    hipMemsetAsync(d_out, 0, (size_t)N_NODES * DIM * sizeof(float), stream);

    // GEMMs (l and r via blockIdx.y); 64KB W + 2x8KB tile double-buffer in LDS
    size_t ldsBytes = (size_t)(DIM * DIM + 2 * 16 * DIM) * sizeof(float);  // 80 KB
    LATTE_gemm<<<dim3(512, 2), 256, ldsBytes, stream>>>(x, Wl, bl, Wr, br, lbuf, rbuf);

    // scores + beta
    LATTE_scores<<<(N_NODES + 7) / 8, 256, 0, stream>>>(lbuf, rbuf, attn, Wbeta,
                                                        bbeta, ssrc, sdst, beta);

    const int* eidx[2] = {ei0, ei1};
    for (int rel = 0; rel < 2; ++rel) {
        const float* ss = ssrc + (size_t)rel * N_NODES * NH;
        const float* sd = sdst + (size_t)rel * N_NODES * NH;
        LATTE_fill<<<(N_NODES * NH + 255) / 256, 256, 0, stream>>>(mbuf, dbuf);
        LATTE_pass1<<<NE / 256, 256, 0, stream>>>(eidx[rel], ss, sd, sharpen, rel, mbuf);
        LATTE_pass2<<<NE / 256, 256, 0, stream>>>(eidx[rel], ss, sd, sharpen, rel, mbuf, dbuf);
        LATTE_pass3<<<NE / 8, 256, 0, stream>>>(eidx[rel], ss, sd, mbuf, dbuf,
                                                lbuf, beta, sharpen, rel, out);
    }

    LATTE_final<<<(N_NODES * DIM + 255) / 256, 256, 0, stream>>>(rbuf, beta, out);
}